// RAGTripletLoss_17540646437126
// MI455X (gfx1250) — compile-verified
//
#include <hip/hip_runtime.h>
#include <stdint.h>

#define N_NODES   200000
#define D_FEAT    128
#define N_EDGES   500000
#define MARGIN_F  0.5f
#define EPS_F     1e-8f

typedef float v2f __attribute__((ext_vector_type(2)));
typedef float v8f __attribute__((ext_vector_type(8)));

static constexpr int TILES           = N_EDGES / 16;   // 31250, exact
static constexpr int WAVES_PER_BLOCK = 8;              // 256 threads = 8 wave32
static constexpr int MAIN_BLOCKS     = (TILES + WAVES_PER_BLOCK - 1) / WAVES_PER_BLOCK; // 3907

__device__ __forceinline__ uint64_t splitmix64(uint64_t x) {
    x += 0x9E3779B97F4A7C15ull;
    x = (x ^ (x >> 30)) * 0xBF58476D1CE4E5B9ull;
    x = (x ^ (x >> 27)) * 0x94D049BB133111EBull;
    return x ^ (x >> 31);
}

// ---- pass 0: zero the per-node class-presence mask (ws is poisoned 0xAA) ----
__global__ void zero_mask_kernel(unsigned* __restrict__ mask) {
    int i = blockIdx.x * blockDim.x + threadIdx.x;
    if (i < N_NODES) mask[i] = 0u;
}

// ---- pass 1: mask[node] |= (1 << class) for both endpoints of every edge ----
// atomicOr is order-independent -> deterministic result.
__global__ void build_mask_kernel(const int* __restrict__ edges,
                                  unsigned* __restrict__ mask) {
    int i = blockIdx.x * blockDim.x + threadIdx.x;
    if (i < N_EDGES) {
        int u = edges[i];
        int v = edges[N_EDGES + i];
        int c = edges[2 * N_EDGES + i];
        if (c >= 0) {
            unsigned bit = 1u << (c & 7);
            atomicOr(&mask[u], bit);
            atomicOr(&mask[v], bit);
        }
    }
}

// ---- pass 2: main triplet kernel. One wave32 handles 16 triplets via WMMA ----
__global__ void __launch_bounds__(256)
triplet_loss_kernel(const float* __restrict__ feats,
                    const int*   __restrict__ edges,
                    const unsigned* __restrict__ mask,
                    float* __restrict__ psum,
                    float* __restrict__ pcnt) {
    const int lane = threadIdx.x & 31;
    const int wave = threadIdx.x >> 5;
    const int tile = blockIdx.x * WAVES_PER_BLOCK + wave;

    float lsum = 0.0f, lcnt = 0.0f;

    if (tile < TILES) {                      // wave-uniform guard: EXEC stays full
        const int row = lane & 15;           // matrix row handled by this lane
        const int hi  = lane >> 4;           // 0: k%4 in {0,1}, 1: k%4 in {2,3}
        const int e   = tile * 16 + row;

        const int ai = edges[e];
        const int pi = edges[N_EDGES + e];
        const int c  = edges[2 * N_EDGES + e] & 7;

        // Deterministic negative: uniform over nodes touching any edge of another
        // class (same candidate set as the reference), via hash + rejection.
        const unsigned notc = ~(1u << c);
        uint64_t s = splitmix64((uint64_t)e ^ 0xA0761D6478BD642Full);
        int ni = 0;
        for (int t = 0; t < 64; ++t) {
            s = splitmix64(s);
            int cand = (int)(s % (uint64_t)N_NODES);
            ni = cand;
            if (mask[cand] & notc) break;    // ~99% accept on first try
        }

        const float* __restrict__ Ar = feats + (size_t)ai * D_FEAT;
        const float* __restrict__ Pr = feats + (size_t)pi * D_FEAT;
        const float* __restrict__ Nr = feats + (size_t)ni * D_FEAT;

        v8f  cap = {};                       // accumulates A x P^T (16x16)
        v8f  can = {};                       // accumulates A x N^T (16x16)
        float naa = 0.0f, npp = 0.0f, nnn = 0.0f;
        const int off = hi * 2;

        // K = 128 in 32 steps of V_WMMA_F32_16X16X4_F32.
        // A-fragment and B^T-fragment have identical per-lane layouts here, so
        // each lane loads one contiguous float2 per matrix per step.
#pragma unroll
        for (int kb = 0; kb < D_FEAT; kb += 4) {
            v2f a = *(const v2f*)(Ar + kb + off);
            v2f p = *(const v2f*)(Pr + kb + off);
            v2f n = *(const v2f*)(Nr + kb + off);
            cap = __builtin_amdgcn_wmma_f32_16x16x4_f32(
                      false, a, false, p, (short)0, cap, false, false);
            can = __builtin_amdgcn_wmma_f32_16x16x4_f32(
                      false, a, false, n, (short)0, can, false, false);
            naa = fmaf(a.x, a.x, fmaf(a.y, a.y, naa));
            npp = fmaf(p.x, p.x, fmaf(p.y, p.y, npp));
            nnn = fmaf(n.x, n.x, fmaf(n.y, n.y, nnn));
        }

        // Combine the two K-halves of the squared norms (lanes j and j+16).
        naa += __shfl_xor(naa, 16, 32);
        npp += __shfl_xor(npp, 16, 32);
        nnn += __shfl_xor(nnn, 16, 32);

        // Diagonal of the 16x16 result: element (i,i) sits at
        //   i in 0..7  -> lane i,      VGPR i
        //   i in 8..15 -> lane i+16,   VGPR i-8
        const bool valid = (lane < 8) || (lane >= 24);
        const int  dl    = (lane < 8) ? lane : (lane - 24);
        float dap = cap[0], dan = can[0];
#pragma unroll
        for (int r = 1; r < 8; ++r) {
            if (dl == r) { dap = cap[r]; dan = can[r]; }
        }

        if (valid) {
            float den_ap = fmaxf(sqrtf(naa) * sqrtf(npp), EPS_F);
            float den_an = fmaxf(sqrtf(naa) * sqrtf(nnn), EPS_F);
            float cs_ap  = dap / den_ap;
            float cs_an  = dan / den_an;
            float loss   = fmaxf(cs_an - cs_ap + MARGIN_F, 0.0f);
            lsum = loss;
            lcnt = (loss > 0.0f) ? 1.0f : 0.0f;
        }
    }

    // Deterministic wave tree reduction.
#pragma unroll
    for (int d = 16; d > 0; d >>= 1) {
        lsum += __shfl_xor(lsum, d, 32);
        lcnt += __shfl_xor(lcnt, d, 32);
    }

    __shared__ float ssum[WAVES_PER_BLOCK];
    __shared__ float scnt[WAVES_PER_BLOCK];
    if (lane == 0) { ssum[wave] = lsum; scnt[wave] = lcnt; }
    __syncthreads();
    if (threadIdx.x == 0) {
        float bs = 0.0f, bc = 0.0f;
#pragma unroll
        for (int w = 0; w < WAVES_PER_BLOCK; ++w) { bs += ssum[w]; bc += scnt[w]; }
        psum[blockIdx.x] = bs;               // fixed slot -> deterministic
        pcnt[blockIdx.x] = bc;
    }
}

// ---- pass 3: single-block fixed-order final reduction -> out = sum / cnt ----
__global__ void final_reduce_kernel(const float* __restrict__ psum,
                                    const float* __restrict__ pcnt,
                                    float* __restrict__ out,
                                    int nblocks) {
    __shared__ float ss[256];
    __shared__ float sc[256];
    float s = 0.0f, c = 0.0f;
    for (int i = threadIdx.x; i < nblocks; i += 256) { s += psum[i]; c += pcnt[i]; }
    ss[threadIdx.x] = s;
    sc[threadIdx.x] = c;
    __syncthreads();
    for (int d = 128; d > 0; d >>= 1) {
        if (threadIdx.x < d) {
            ss[threadIdx.x] += ss[threadIdx.x + d];
            sc[threadIdx.x] += sc[threadIdx.x + d];
        }
        __syncthreads();
    }
    if (threadIdx.x == 0) out[0] = ss[0] / sc[0];
}

extern "C" void kernel_launch(void* const* d_in, const int* in_sizes, int n_in,
                              void* d_out, int out_size, void* d_ws, size_t ws_size,
                              hipStream_t stream) {
    (void)in_sizes; (void)n_in; (void)out_size; (void)ws_size;
    const float* feats = (const float*)d_in[0];     // [N_NODES, 128] f32
    const int*   edges = (const int*)d_in[1];       // [3, N_EDGES]  i32

    unsigned* mask = (unsigned*)d_ws;                                    // 800 KB
    float*    psum = (float*)((char*)d_ws + (size_t)N_NODES * sizeof(unsigned));
    float*    pcnt = psum + MAIN_BLOCKS;

    zero_mask_kernel<<<(N_NODES + 255) / 256, 256, 0, stream>>>(mask);
    build_mask_kernel<<<(N_EDGES + 255) / 256, 256, 0, stream>>>(edges, mask);
    triplet_loss_kernel<<<MAIN_BLOCKS, 256, 0, stream>>>(feats, edges, mask, psum, pcnt);
    final_reduce_kernel<<<1, 256, 0, stream>>>(psum, pcnt, (float*)d_out, MAIN_BLOCKS);
}